// PointNetSAModule_86260123173794
// MI455X (gfx1250) — compile-verified
//
#include <hip/hip_runtime.h>
#include <hip/hip_bf16.h>

// ---- problem constants (match reference) ----
#define B_    16
#define N_    4096
#define CIN_  64
#define M_    1024
#define KNB_  32
#define C0_   67                 // 3 coords + 64 features
#define C0P_  68                 // padded to multiple of 4 for WMMA K-steps
#define NC_   (B_*M_*KNB_)       // 524288 columns
#define RADIUS2_ 0.04f
#define EPS_  1e-5f

typedef float v2f __attribute__((ext_vector_type(2)));
typedef float v8f __attribute__((ext_vector_type(8)));

// =====================================================================
// 1) Farthest point sampling: one 1024-thread block per batch.
//    Coords live in LDS; per-thread min-dist in registers; argmax via
//    wave32 shuffle butterfly + one cross-wave LDS stage (2 barriers/step).
// =====================================================================
__global__ __launch_bounds__(1024) void fps_kernel(const float* __restrict__ coords,
                                                   float* __restrict__ centers)
{
    __shared__ float sX[N_], sY[N_], sZ[N_];
    __shared__ int   sC[M_];
    __shared__ float sRv[32];
    __shared__ int   sRi[32];
    __shared__ int   sLast;

    const int b   = blockIdx.x;
    const int tid = threadIdx.x;
    const float* cb = coords + (size_t)b * 3 * N_;
    for (int i = tid; i < N_; i += 1024) {
        sX[i] = cb[i];
        sY[i] = cb[N_ + i];
        sZ[i] = cb[2 * N_ + i];
    }
    if (tid == 0) sLast = 0;
    __syncthreads();

    const int p0 = tid * 4;
    float mind[4] = {1e10f, 1e10f, 1e10f, 1e10f};
    float px[4], py[4], pz[4];
    #pragma unroll
    for (int j = 0; j < 4; ++j) { px[j] = sX[p0+j]; py[j] = sY[p0+j]; pz[j] = sZ[p0+j]; }

    const int lane = tid & 31;
    const int wave = tid >> 5;

    for (int s = 0; s < M_; ++s) {
        const int L = sLast;
        if (tid == 0) sC[s] = L;
        const float lx = sX[L], ly = sY[L], lz = sZ[L];

        float bv = -1.0f; int bi = 0;
        #pragma unroll
        for (int j = 0; j < 4; ++j) {
            const float dx = px[j]-lx, dy = py[j]-ly, dz = pz[j]-lz;
            const float d  = dx*dx + dy*dy + dz*dz;
            const float mn = fminf(mind[j], d);
            mind[j] = mn;
            if (mn > bv) { bv = mn; bi = p0 + j; }   // ascending j keeps first max
        }
        // wave32 argmax reduce (tie -> lowest index, matching jnp.argmax)
        #pragma unroll
        for (int off = 16; off >= 1; off >>= 1) {
            const float ov = __shfl_xor(bv, off, 32);
            const int   oi = __shfl_xor(bi, off, 32);
            if (ov > bv || (ov == bv && oi < bi)) { bv = ov; bi = oi; }
        }
        if (lane == 0) { sRv[wave] = bv; sRi[wave] = bi; }
        __syncthreads();
        if (wave == 0) {
            float v = sRv[lane]; int i2 = sRi[lane];
            #pragma unroll
            for (int off = 16; off >= 1; off >>= 1) {
                const float ov = __shfl_xor(v, off, 32);
                const int   oi = __shfl_xor(i2, off, 32);
                if (ov > v || (ov == v && oi < i2)) { v = ov; i2 = oi; }
            }
            if (lane == 0) sLast = i2;
        }
        __syncthreads();
    }

    // centers (B,3,M) -> second region of d_out
    if (tid < M_) {
        const int ci = sC[tid];
        float* cc = centers + (size_t)b * 3 * M_;
        cc[tid]          = sX[ci];
        cc[M_ + tid]     = sY[ci];
        cc[2 * M_ + tid] = sZ[ci];
    }
}

// =====================================================================
// 2) Ball query: one wave32 per center; ballot + prefix-popc gives the
//    in-index-order "first K inside radius" selection; uniform early exit.
// =====================================================================
__global__ void ballquery_kernel(const float* __restrict__ coords,
                                 const float* __restrict__ centers,
                                 int* __restrict__ nidx)
{
    const int gtid = blockIdx.x * blockDim.x + threadIdx.x;
    const int wid  = gtid >> 5;          // center id
    const int lane = gtid & 31;
    const int b = wid / M_;
    const int m = wid % M_;

    const float* cc = centers + (size_t)b * 3 * M_;
    const float cx = cc[m], cy = cc[M_ + m], cz = cc[2 * M_ + m];
    const float* pb = coords + (size_t)b * 3 * N_;
    int* out = nidx + (size_t)wid * KNB_;

    int cnt = 0, first = 0;
    bool haveFirst = false;
    for (int chunk = 0; chunk < (N_ / 32) && cnt < KNB_; ++chunk) {
        const int j = chunk * 32 + lane;
        const float dx = pb[j] - cx, dy = pb[N_ + j] - cy, dz = pb[2*N_ + j] - cz;
        const bool pred = (dx*dx + dy*dy + dz*dz) < RADIUS2_;
        const unsigned mask = (unsigned)__ballot(pred);
        if (!haveFirst && mask) { first = chunk * 32 + (__ffs(mask) - 1); haveFirst = true; }
        const int off = __popc(mask & ((1u << lane) - 1u));
        if (pred && (cnt + off) < KNB_) out[cnt + off] = j;
        cnt += __popc(mask);
    }
    if (cnt > KNB_) cnt = KNB_;
    const int pad = haveFirst ? first : 0;
    if (lane >= cnt) out[lane] = pad;    // KNB_ == warp width == 32
}

// =====================================================================
// 3) Gather + concat into padded activation: x0[c][col], c<3 = centered
//    coords, 3..66 = features, c==67 = zero padding row.
// =====================================================================
__global__ void gather_kernel(const float* __restrict__ features,
                              const float* __restrict__ coords,
                              const float* __restrict__ centers,
                              const int* __restrict__ nidx,
                              float* __restrict__ x0)
{
    const size_t total = (size_t)C0P_ * NC_;
    for (size_t i = blockIdx.x * (size_t)blockDim.x + threadIdx.x; i < total;
         i += (size_t)gridDim.x * blockDim.x) {
        const int c   = (int)(i / NC_);
        const int col = (int)(i % NC_);
        const int b   = col / (M_ * KNB_);
        const int rem = col % (M_ * KNB_);
        const int m   = rem / KNB_;
        float v = 0.0f;
        if (c < C0_) {
            const int pt = nidx[col];
            if (c < 3) {
                v = coords[((size_t)b * 3 + c) * N_ + pt]
                  - centers[((size_t)b * 3 + c) * M_ + m];
            } else {
                v = features[((size_t)b * CIN_ + (c - 3)) * N_ + pt];
            }
        }
        x0[(size_t)c * NC_ + col] = v;
    }
}

// Pack w1 (64 x 67) into zero-padded (64 x 68)
__global__ void pack_w1_kernel(const float* __restrict__ w, float* __restrict__ wp)
{
    const int i = blockIdx.x * blockDim.x + threadIdx.x;
    if (i >= 64 * C0P_) return;
    const int r = i / C0P_, c = i % C0P_;
    wp[i] = (c < C0_) ? w[r * C0_ + c] : 0.0f;
}

// =====================================================================
// 4) GEMM  Y = W (COUT x CINP) @ X (CINP x NC) + bias, via
//    V_WMMA_F32_16X16X4_F32. Fully compile-time: K loop unrolled,
//    branch-free (padding instead of guards), A panel held in registers
//    (CINP/4 x v2f) and reused across CTILES column tiles per wave.
//    Grids sized exactly => EXEC all-1s at every WMMA.
//    A 16x4 layout: lanes 0-15 row M=lane {K0,K1}; lanes 16-31 {K2,K3}.
//    B 4x16 layout: symmetric (col N=lane&15). C/D: VGPR j -> M=j / j+8.
// =====================================================================
template<int CINP, int COUT, int CTILES>
__global__ __launch_bounds__(256) void gemm_wmma_kernel(
        const float* __restrict__ W,      // COUT x CINP (CINP % 4 == 0)
        const float* __restrict__ bias,   // COUT
        const float* __restrict__ X,      // CINP rows x NC_ cols
        float* __restrict__ Y)            // COUT rows x NC_ cols
{
    constexpr int KSTEPS = CINP / 4;
    constexpr int mTiles = COUT / 16;

    const int lane  = threadIdx.x & 31;
    const int gwave = blockIdx.x * (blockDim.x >> 5) + (threadIdx.x >> 5);
    const int tileM = gwave % mTiles;
    const int tileN = gwave / mTiles;

    const int rowA = tileM * 16 + (lane & 15);
    const int klo  = (lane >> 4) * 2;               // 0 or 2
    const int col0 = tileN * (16 * CTILES) + (lane & 15);

    // ---- A panel: 16 x CINP weight tile, resident in registers ----
    v2f a[KSTEPS];
    const float* wr = W + rowA * CINP + klo;        // 8B-aligned (CINP even, klo even)
    #pragma unroll
    for (int ks = 0; ks < KSTEPS; ++ks)
        a[ks] = *(const v2f*)(wr + ks * 4);

    const int rbase = tileM * 16 + ((lane >> 4) << 3);
    float bv[8];
    #pragma unroll
    for (int j = 0; j < 8; ++j) bv[j] = bias[rbase + j];

    #pragma unroll
    for (int t = 0; t < CTILES; ++t) {
        const int col = col0 + 16 * t;
        const float* xc = X + (size_t)klo * NC_ + col;
        v8f acc = {};
        #pragma unroll
        for (int ks = 0; ks < KSTEPS; ++ks) {
            v2f bb;
            bb.x = xc[(size_t)(ks * 4) * NC_];
            bb.y = xc[(size_t)(ks * 4) * NC_ + NC_];
            acc = __builtin_amdgcn_wmma_f32_16x16x4_f32(
                      false, a[ks], false, bb, (short)0, acc, false, false);
        }
        #pragma unroll
        for (int j = 0; j < 8; ++j)
            Y[(size_t)(rbase + j) * NC_ + col] = acc[j] + bv[j];
    }
}

// =====================================================================
// 5) BN stats: deterministic two-stage (per-block partials, serial final)
// =====================================================================
#define STAT_CHUNKS_ 32
__global__ void stats_kernel(const float* __restrict__ Y,
                             float* __restrict__ psum, float* __restrict__ psq)
{
    const int CH = NC_ / STAT_CHUNKS_;            // 16384
    const int ch    = blockIdx.x / STAT_CHUNKS_;
    const int chunk = blockIdx.x % STAT_CHUNKS_;
    const float* p = Y + (size_t)ch * NC_ + (size_t)chunk * CH;
    float s = 0.f, q = 0.f;
    for (int i = threadIdx.x; i < CH; i += blockDim.x) { const float v = p[i]; s += v; q += v * v; }
    #pragma unroll
    for (int off = 16; off >= 1; off >>= 1) { s += __shfl_xor(s, off, 32); q += __shfl_xor(q, off, 32); }
    __shared__ float ss[8], sq[8];
    const int lane = threadIdx.x & 31, wv = threadIdx.x >> 5;
    if (lane == 0) { ss[wv] = s; sq[wv] = q; }
    __syncthreads();
    if (threadIdx.x == 0) {
        float S = 0.f, Q = 0.f;
        for (int w = 0; w < 8; ++w) { S += ss[w]; Q += sq[w]; }
        psum[ch * STAT_CHUNKS_ + chunk] = S;
        psq [ch * STAT_CHUNKS_ + chunk] = Q;
    }
}

__global__ void finalize_stats_kernel(const float* __restrict__ psum,
                                      const float* __restrict__ psq,
                                      float* __restrict__ mean,
                                      float* __restrict__ rstd, int Cout)
{
    const int c = threadIdx.x;
    if (c < Cout) {
        float S = 0.f, Q = 0.f;
        for (int i = 0; i < STAT_CHUNKS_; ++i) { S += psum[c*STAT_CHUNKS_+i]; Q += psq[c*STAT_CHUNKS_+i]; }
        const float cnt = (float)NC_;
        const float mu  = S / cnt;
        const float var = Q / cnt - mu * mu;
        mean[c] = mu;
        rstd[c] = rsqrtf(var + EPS_);
    }
}

__global__ void norm_relu_kernel(float* __restrict__ Y,
                                 const float* __restrict__ mean,
                                 const float* __restrict__ rstd,
                                 const float* __restrict__ g,
                                 const float* __restrict__ bt, int Cout)
{
    const size_t total = (size_t)Cout * NC_;
    for (size_t i = blockIdx.x * (size_t)blockDim.x + threadIdx.x; i < total;
         i += (size_t)gridDim.x * blockDim.x) {
        const int ch = (int)(i / NC_);
        float v = Y[i];
        v = (v - mean[ch]) * rstd[ch] * g[ch] + bt[ch];
        Y[i] = v > 0.f ? v : 0.f;
    }
}

// =====================================================================
// 6) Final: normalize + ReLU + max over K, write out (B,128,M)
// =====================================================================
__global__ void norm_maxpool_kernel(const float* __restrict__ Y,
                                    const float* __restrict__ mean,
                                    const float* __restrict__ rstd,
                                    const float* __restrict__ g,
                                    const float* __restrict__ bt,
                                    float* __restrict__ out)
{
    const int i = blockIdx.x * blockDim.x + threadIdx.x;     // b*128*M + ch*M + m
    if (i >= B_ * 128 * M_) return;
    const int b   = i / (128 * M_);
    const int rem = i % (128 * M_);
    const int ch  = rem / M_;
    const int m   = rem % M_;
    const float* p = Y + (size_t)ch * NC_ + ((size_t)(b * M_ + m)) * KNB_;
    const float mu = mean[ch], rs = rstd[ch], gg = g[ch], bb = bt[ch];
    float mx = -1e30f;
    #pragma unroll 8
    for (int k = 0; k < KNB_; ++k) {
        float v = (p[k] - mu) * rs * gg + bb;
        v = v > 0.f ? v : 0.f;
        mx = fmaxf(mx, v);
    }
    out[i] = mx;
}

// =====================================================================
// Launcher
// =====================================================================
extern "C" void kernel_launch(void* const* d_in, const int* in_sizes, int n_in,
                              void* d_out, int out_size, void* d_ws, size_t ws_size,
                              hipStream_t stream) {
    const float* features = (const float*)d_in[0];
    const float* coords   = (const float*)d_in[1];
    const float* w1 = (const float*)d_in[2];
    const float* b1 = (const float*)d_in[3];
    const float* g1 = (const float*)d_in[4];
    const float* bt1= (const float*)d_in[5];
    const float* w2 = (const float*)d_in[6];
    const float* b2 = (const float*)d_in[7];
    const float* g2 = (const float*)d_in[8];
    const float* bt2= (const float*)d_in[9];
    const float* w3 = (const float*)d_in[10];
    const float* b3 = (const float*)d_in[11];
    const float* g3 = (const float*)d_in[12];
    const float* bt3= (const float*)d_in[13];

    float* out     = (float*)d_out;
    float* centers = out + (size_t)B_ * 128 * M_;   // tuple tail: (B,3,M)

    // ---- workspace carve-up ----
    char* ws = (char*)d_ws;
    size_t off = 0;
    auto carve = [&](size_t bytes) { void* p = ws + off; off = (off + bytes + 255) & ~(size_t)255; return p; };
    int*   nidx = (int*)  carve((size_t)NC_ * sizeof(int));
    float* psum = (float*)carve(128 * STAT_CHUNKS_ * sizeof(float));
    float* psq  = (float*)carve(128 * STAT_CHUNKS_ * sizeof(float));
    float* mean = (float*)carve(128 * sizeof(float));
    float* rstd = (float*)carve(128 * sizeof(float));
    float* w1p  = (float*)carve((size_t)64 * C0P_ * sizeof(float));
    float* bufX = (float*)carve((size_t)C0P_ * NC_ * sizeof(float));  // x0, later y2
    float* bufY = (float*)carve((size_t)128  * NC_ * sizeof(float));  // y1 / y3
    (void)ws_size; (void)in_sizes; (void)n_in; (void)out_size;

    // 1) FPS -> centers
    fps_kernel<<<B_, 1024, 0, stream>>>(coords, centers);

    // 2) ball query -> nidx
    ballquery_kernel<<<(B_ * M_ * 32) / 256, 256, 0, stream>>>(coords, centers, nidx);

    // 3) gather/concat -> bufX (68 x NC, row 67 zero), pack padded w1
    gather_kernel<<<32768, 256, 0, stream>>>(features, coords, centers, nidx, bufX);
    pack_w1_kernel<<<(64 * C0P_ + 255) / 256, 256, 0, stream>>>(w1, w1p);

    // ---- layer 1: 68(padded) -> 64 ----
    // waves = mTiles(4) * colGroups(NC/64 = 8192) = 32768 -> 4096 blocks
    gemm_wmma_kernel<C0P_, 64, 4><<<4096, 256, 0, stream>>>(w1p, b1, bufX, bufY);
    stats_kernel<<<64 * STAT_CHUNKS_, 256, 0, stream>>>(bufY, psum, psq);
    finalize_stats_kernel<<<1, 128, 0, stream>>>(psum, psq, mean, rstd, 64);
    norm_relu_kernel<<<32768, 256, 0, stream>>>(bufY, mean, rstd, g1, bt1, 64);

    // ---- layer 2: 64 -> 64 (bufY -> bufX) ----
    gemm_wmma_kernel<64, 64, 4><<<4096, 256, 0, stream>>>(w2, b2, bufY, bufX);
    stats_kernel<<<64 * STAT_CHUNKS_, 256, 0, stream>>>(bufX, psum, psq);
    finalize_stats_kernel<<<1, 128, 0, stream>>>(psum, psq, mean, rstd, 64);
    norm_relu_kernel<<<32768, 256, 0, stream>>>(bufX, mean, rstd, g2, bt2, 64);

    // ---- layer 3: 64 -> 128 (bufX -> bufY) ----
    // waves = mTiles(8) * 8192 = 65536 -> 8192 blocks
    gemm_wmma_kernel<64, 128, 4><<<8192, 256, 0, stream>>>(w3, b3, bufX, bufY);
    stats_kernel<<<128 * STAT_CHUNKS_, 256, 0, stream>>>(bufY, psum, psq);
    finalize_stats_kernel<<<1, 128, 0, stream>>>(psum, psq, mean, rstd, 128);

    // ---- norm + relu + max over K -> out ----
    norm_maxpool_kernel<<<(B_ * 128 * M_) / 256, 256, 0, stream>>>(bufY, mean, rstd, g3, bt3, out);
}